// HamiltonProductAMX_19670950216617
// MI455X (gfx1250) — compile-verified
//
#include <hip/hip_runtime.h>
#include <hip/hip_bf16.h>

// CDNA5 / gfx1250 Hamilton product via V_WMMA_F32_16X16X4_F32 (full f32 precision).
//
// Math: r_n = L(q_n) x_n = sum_{t=0..3} x_n[t] * (Pi_t q_n)
// Flatten K over (t,j): D[c][n] = sum_kk A[c][kk] * B[kk][n]
//   A[c][(t,j)] = sign_t[c] iff j == perm_t[c]   (constant +-1, rows 4..15 zero)
//   B[(t,j)][n] = q_n[j] * x_n[t]                (per-lane outer product)
// K = 16 -> 4 chained 16x16x4 WMMAs, 16 quaternions per wave.
//
// A-matrix (16x4 f32, 2 VGPRs): lanes 0-15 hold K=0/1, lanes 16-31 hold K=2/3 (row = lane&15).
// B-matrix (4x16 f32, 2 VGPRs): lane's column = lane&15; lanes 0-15 hold K=0/1, 16-31 K=2/3.
// C/D (16x16 f32, 8 VGPRs): VGPR r = row M=r for lanes 0-15 -> acc[0..3] at lane n is r_n.

typedef __attribute__((ext_vector_type(2))) float v2f;
typedef __attribute__((ext_vector_type(8))) float v8f;

__global__ __launch_bounds__(256) void hamilton_wmma_f32_kernel(
    const float* __restrict__ q,
    const float* __restrict__ x,
    float* __restrict__ out,
    int nquat)
{
    const int tid  = blockIdx.x * blockDim.x + threadIdx.x;
    const int wave = tid >> 5;          // one 16-quat tile per wave32
    const int lane = threadIdx.x & 31;
    const int row  = lane & 15;         // A-row / B-column index
    const int half = lane >> 4;         // 0: K=0,1   1: K=2,3

    // Signed permutations Pi_t (columns of L(q)):
    //   t0: [ w,  i,  j,  k]  perm 0123  signs ++++
    //   t1: [-i,  w,  k, -j]  perm 1032  signs -++-
    //   t2: [-j, -k,  w,  i]  perm 2301  signs --++
    //   t3: [-k,  j, -i,  w]  perm 3210  signs -+-+
    const unsigned perm[4] = {0xE4u, 0xB1u, 0x4Eu, 0x1Bu}; // 2 bits per row, row0 in LSBs
    const unsigned sneg[4] = {0x0u,  0x9u,  0x3u,  0x5u};  // bit c set -> negative

    // Per-lane constant A entries for the two K slots this lane owns, per step t.
    float a0[4], a1[4];
#pragma unroll
    for (int t = 0; t < 4; ++t) {
        const unsigned p = (perm[t] >> (2 * row)) & 3u;
        const float    s = ((sneg[t] >> row) & 1u) ? -1.0f : 1.0f;
        const bool  live = (row < 4);
        a0[t] = (live && p == (unsigned)(2 * half))     ? s : 0.0f;
        a1[t] = (live && p == (unsigned)(2 * half + 1)) ? s : 0.0f;
    }

    // Both wave halves load the same 16 quats (same cachelines -> no extra HBM traffic).
    const int idx_raw = wave * 16 + row;
    const int idx     = (idx_raw < nquat) ? idx_raw : (nquat - 1); // clamp: EXEC must stay all-ones for WMMA
    const float4 qv = reinterpret_cast<const float4*>(q)[idx];     // global_load_b128
    const float4 xv = reinterpret_cast<const float4*>(x)[idx];     // global_load_b128

    // This lane's q components for its K slots: j = 2*half, 2*half+1.
    const float qa = half ? qv.z : qv.x;
    const float qb = half ? qv.w : qv.y;
    const float xt[4] = {xv.x, xv.y, xv.z, xv.w};

    v8f acc = {};
#pragma unroll
    for (int t = 0; t < 4; ++t) {
        v2f a; a[0] = a0[t];      a[1] = a1[t];
        v2f b; b[0] = qa * xt[t]; b[1] = qb * xt[t];   // B[(t,j)][n] = q_n[j]*x_n[t]
        // 8 args: (neg_a, A, neg_b, B, c_mod, C, reuse_a, reuse_b)
        acc = __builtin_amdgcn_wmma_f32_16x16x4_f32(
            /*neg_a=*/false, a, /*neg_b=*/false, b,
            /*c_mod=*/(short)0, acc, /*reuse_a=*/false, /*reuse_b=*/false);
    }

    // Lanes 0-15 hold D rows 0..3 at column N=lane -> one full result quaternion each.
    if (half == 0 && idx_raw < nquat) {
        float4 r;
        r.x = acc[0]; r.y = acc[1]; r.z = acc[2]; r.w = acc[3];
        reinterpret_cast<float4*>(out)[idx_raw] = r;               // global_store_b128
    }
}

extern "C" void kernel_launch(void* const* d_in, const int* in_sizes, int n_in,
                              void* d_out, int out_size, void* d_ws, size_t ws_size,
                              hipStream_t stream) {
    const float* q = (const float*)d_in[0];
    const float* x = (const float*)d_in[1];
    float* out = (float*)d_out;

    const int nquat = in_sizes[0] / 4;          // 8,388,608
    // 256 threads = 8 wave32s, 16 quats per wave -> 128 quats per block.
    const int blocks = (nquat + 127) / 128;
    hamilton_wmma_f32_kernel<<<blocks, 256, 0, stream>>>(q, x, out, nquat);
}